// SimpleEVGNN_46815143527094
// MI455X (gfx1250) — compile-verified
//
#include <hip/hip_runtime.h>
#include <hip/hip_bf16.h>

typedef __attribute__((ext_vector_type(2))) float v2f;
typedef __attribute__((ext_vector_type(8))) float v8f;

#define N_NODES 100000
#define N_EDGES 1600000
#define IN_CH   128
#define HID_CH  128

// ---------------------------------------------------------------------------
// Degree computation: deg starts at 1.0 (self loop), +1 per incoming edge,
// then replaced in-place by rsqrt(deg).
// ---------------------------------------------------------------------------
__global__ void k_init_deg(float* __restrict__ deg, int n) {
    int i = blockIdx.x * blockDim.x + threadIdx.x;
    if (i < n) deg[i] = 1.0f;
}

__global__ void k_count_deg(const int* __restrict__ dst, float* __restrict__ deg, int nE) {
    int e = blockIdx.x * blockDim.x + threadIdx.x;
    if (e < nE) atomicAdd(&deg[dst[e]], 1.0f);
}

__global__ void k_rsqrt_inplace(float* __restrict__ deg, int n) {
    int i = blockIdx.x * blockDim.x + threadIdx.x;
    if (i < n) deg[i] = rsqrtf(deg[i]);
}

// ---------------------------------------------------------------------------
// Layer-1 GEMM: XW[100000,128] = X[100000,128] @ W[128,128], fp32 WMMA.
// One wave per 16-row strip: the A fragment is loaded ONCE per k-step and
// reused across all 8 N-tiles (8 accumulators, 64 C VGPRs). 8x less X traffic
// than tile-per-wave; 8 WMMA per 9 loads.
// ---------------------------------------------------------------------------
__global__ __launch_bounds__(256) void k_gemm_xw_wmma(
    const float* __restrict__ X, const float* __restrict__ W,
    float* __restrict__ XW, int M) {
    const int wavesPerBlock = blockDim.x >> 5;
    const int waveId = blockIdx.x * wavesPerBlock + (threadIdx.x >> 5);
    const int lane = threadIdx.x & 31;
    const int m0 = waveId * 16;
    if (m0 >= M) return;                             // wave-uniform guard
    const int half = lane >> 4;                      // 0: K+0/1, 1: K+2/3
    const int l = lane & 15;

    const float* xrow = X + (size_t)(m0 + l) * IN_CH;
    v8f acc[8] = {};
#pragma unroll 2
    for (int k = 0; k < IN_CH; k += 4) {
        const int ka = k + 2 * half;
        v2f a;
        a.x = xrow[ka + 0];                          // A(M=l, K=ka)
        a.y = xrow[ka + 1];                          // A(M=l, K=ka+1)
        const float* w0 = W + (size_t)ka * HID_CH + l;
#pragma unroll
        for (int nt = 0; nt < 8; ++nt) {
            v2f b;
            b.x = w0[nt * 16];                       // B(K=ka,   N=nt*16+l)
            b.y = w0[HID_CH + nt * 16];              // B(K=ka+1, N=nt*16+l)
            acc[nt] = __builtin_amdgcn_wmma_f32_16x16x4_f32(
                /*neg_a=*/false, a, /*neg_b=*/false, b,
                /*c_mod=*/(short)0, acc[nt], /*reuse_a=*/false, /*reuse_b=*/false);
        }
    }
    // D layout: VGPR r -> row m0 + r + 8*half, col nt*16 + (lane&15)
#pragma unroll
    for (int r = 0; r < 8; ++r) {
        float* orow = XW + (size_t)(m0 + r + 8 * half) * HID_CH + l;
#pragma unroll
        for (int nt = 0; nt < 8; ++nt) {
            orow[nt * 16] = acc[nt][r];
        }
    }
}

// ---------------------------------------------------------------------------
// Self-loop contribution initializes the accumulator:
// acc[i][c] = xw[i][c] * dinv[i]^2
// ---------------------------------------------------------------------------
__global__ void k_self_loop_init(const float* __restrict__ xw,
                                 const float* __restrict__ dinv,
                                 float* __restrict__ acc, int n) {
    int idx = blockIdx.x * blockDim.x + threadIdx.x;   // over n*128
    if (idx < n * HID_CH) {
        int i = idx >> 7;
        float di = dinv[i];
        acc[idx] = xw[idx] * di * di;
    }
}

// ---------------------------------------------------------------------------
// Edge scatter (layer 1): one wave per edge. Edge index is forced into an
// SGPR so src/dst/dinv loads scalarize (4 SMEM loads per edge instead of 128
// VMEM loads). Each lane gathers float4 of xw[src] (coalesced 512B) and does
// 4 fp32 atomics into the L2-resident accumulator.
// ---------------------------------------------------------------------------
__global__ __launch_bounds__(256) void k_edge_scatter1(
    const float* __restrict__ xw, const int* __restrict__ src,
    const int* __restrict__ dst, const float* __restrict__ dinv,
    float* __restrict__ acc, int nE) {
    int gid = blockIdx.x * blockDim.x + threadIdx.x;
    int e = __builtin_amdgcn_readfirstlane(gid >> 5);  // wave-uniform -> SGPR
    int lane = threadIdx.x & 31;
    if (e >= nE) return;
    int s = src[e], d = dst[e];                        // scalar loads
    float norm = dinv[s] * dinv[d];                    // scalar loads
    const float4* xs = (const float4*)(xw + (size_t)s * HID_CH);
    float4 v = xs[lane];
    float* out = acc + (size_t)d * HID_CH + lane * 4;
    atomicAdd(out + 0, v.x * norm);
    atomicAdd(out + 1, v.y * norm);
    atomicAdd(out + 2, v.z * norm);
    atomicAdd(out + 3, v.w * norm);
}

// ---------------------------------------------------------------------------
// h = relu(acc + b1), in place.
// ---------------------------------------------------------------------------
__global__ void k_relu_bias(float* __restrict__ acc, const float* __restrict__ b1, int n) {
    int idx = blockIdx.x * blockDim.x + threadIdx.x;   // over n*128
    if (idx < n * HID_CH) {
        int c = idx & (HID_CH - 1);
        acc[idx] = fmaxf(acc[idx] + b1[c], 0.0f);
    }
}

// ---------------------------------------------------------------------------
// Layer-2 GEMV: hw[i] = dot(h[i,:], W2[:,0]). One wave per node, float4 per
// lane, wave32 xor-shuffle reduction.
// ---------------------------------------------------------------------------
__global__ __launch_bounds__(256) void k_gemv_w2(
    const float* __restrict__ h, const float* __restrict__ W2,
    float* __restrict__ hw, int n) {
    int gid = blockIdx.x * blockDim.x + threadIdx.x;
    int node = __builtin_amdgcn_readfirstlane(gid >> 5);
    int lane = threadIdx.x & 31;
    if (node >= n) return;                           // wave-uniform
    const float4* hr = (const float4*)(h + (size_t)node * HID_CH);
    const float4* wr = (const float4*)W2;
    float4 a = hr[lane];
    float4 b = wr[lane];
    float sum = a.x * b.x + a.y * b.y + a.z * b.z + a.w * b.w;
#pragma unroll
    for (int off = 16; off >= 1; off >>= 1)
        sum += __shfl_xor(sum, off, 32);
    if (lane == 0) hw[node] = sum;
}

// ---------------------------------------------------------------------------
// Layer-2 output init (self loop + bias): out[i] = hw[i]*dinv[i]^2 + b2[0]
// ---------------------------------------------------------------------------
__global__ void k_out_init2(const float* __restrict__ hw,
                            const float* __restrict__ dinv,
                            const float* __restrict__ b2,
                            float* __restrict__ out, int n) {
    int i = blockIdx.x * blockDim.x + threadIdx.x;
    if (i < n) {
        float di = dinv[i];
        out[i] = hw[i] * di * di + b2[0];
    }
}

// ---------------------------------------------------------------------------
// Layer-2 edge scatter: one thread per edge, scalar fp32 atomic.
// ---------------------------------------------------------------------------
__global__ void k_edge_scatter2(const float* __restrict__ hw,
                                const int* __restrict__ src,
                                const int* __restrict__ dst,
                                const float* __restrict__ dinv,
                                float* __restrict__ out, int nE) {
    int e = blockIdx.x * blockDim.x + threadIdx.x;
    if (e < nE) {
        int s = src[e], d = dst[e];
        atomicAdd(&out[d], hw[s] * dinv[s] * dinv[d]);
    }
}

extern "C" void kernel_launch(void* const* d_in, const int* in_sizes, int n_in,
                              void* d_out, int out_size, void* d_ws, size_t ws_size,
                              hipStream_t stream) {
    (void)in_sizes; (void)n_in; (void)out_size; (void)ws_size;

    const float* x  = (const float*)d_in[0];              // [N, 128]
    const int*   ei = (const int*)d_in[1];                // [2, E]
    const float* W1 = (const float*)d_in[2];              // [128, 128]
    const float* b1 = (const float*)d_in[3];              // [128]
    const float* W2 = (const float*)d_in[4];              // [128, 1]
    const float* b2 = (const float*)d_in[5];              // [1]
    float* out = (float*)d_out;                           // [N]

    const int* src = ei;                                  // row 0
    const int* dst = ei + N_EDGES;                        // row 1

    // Workspace carve-up (all offsets 256B-aligned)
    char* ws = (char*)d_ws;
    size_t off = 0;
    auto carve = [&](size_t bytes) {
        char* p = ws + off;
        off += (bytes + 255) & ~(size_t)255;
        return p;
    };
    float* dinv = (float*)carve((size_t)N_NODES * sizeof(float));          // deg -> dinv
    float* xw   = (float*)carve((size_t)N_NODES * HID_CH * sizeof(float)); // 51.2 MB
    float* acc  = (float*)carve((size_t)N_NODES * HID_CH * sizeof(float)); // 51.2 MB (-> h)
    float* hw   = (float*)carve((size_t)N_NODES * sizeof(float));

    const int TB = 256;
    const int nNodeBlk = (N_NODES + TB - 1) / TB;                    // 391
    const int nEdgeBlk = (N_EDGES + TB - 1) / TB;                    // 6250
    const int nElemBlk = (N_NODES * HID_CH + TB - 1) / TB;           // 50000
    const int nEdgeWv  = ((size_t)N_EDGES * 32 + TB - 1) / TB;       // 200000
    const int nNodeWv  = ((size_t)N_NODES * 32 + TB - 1) / TB;       // 12500
    const int nGemmWaves = N_NODES / 16;                             // 6250 strips
    const int nGemmBlk = (nGemmWaves + (TB / 32) - 1) / (TB / 32);   // 782

    // Degrees with self-loops -> dinv
    k_init_deg<<<nNodeBlk, TB, 0, stream>>>(dinv, N_NODES);
    k_count_deg<<<nEdgeBlk, TB, 0, stream>>>(dst, dinv, N_EDGES);
    k_rsqrt_inplace<<<nNodeBlk, TB, 0, stream>>>(dinv, N_NODES);

    // Layer 1
    k_gemm_xw_wmma<<<nGemmBlk, TB, 0, stream>>>(x, W1, xw, N_NODES);
    k_self_loop_init<<<nElemBlk, TB, 0, stream>>>(xw, dinv, acc, N_NODES);
    k_edge_scatter1<<<nEdgeWv, TB, 0, stream>>>(xw, src, dst, dinv, acc, N_EDGES);
    k_relu_bias<<<nElemBlk, TB, 0, stream>>>(acc, b1, N_NODES);

    // Layer 2
    k_gemv_w2<<<nNodeWv, TB, 0, stream>>>(acc, W2, hw, N_NODES);
    k_out_init2<<<nNodeBlk, TB, 0, stream>>>(hw, dinv, b2, out, N_NODES);
    k_edge_scatter2<<<nEdgeBlk, TB, 0, stream>>>(hw, src, dst, dinv, out, N_EDGES);
}